// RecurrentAttention_64355789963726
// MI455X (gfx1250) — compile-verified
//
#include <hip/hip_runtime.h>
#include <cstdint>

// Dims: T=512, B=64, IN=H=OUT=512, gates=2048, K(recurrent GEMMs)=1024.
typedef __attribute__((ext_vector_type(2))) float v2f;
typedef __attribute__((ext_vector_type(8))) float v8f;

__device__ __forceinline__ float sigf(float x) { return 1.0f / (1.0f + expf(-x)); }

// workspace float offsets
#define WS_WTS      0u        // 512*64          softmax-over-T weights (constant across steps)
#define WS_CTX      32768u    // 64*512          attention context (constant across steps)
#define WS_GATESC   65536u    // 64*2048         c-part of gates + biases
#define WS_ATTNC    196608u   // 64*512          c-part of attn linear + bias
#define WS_BIAS0    229376u   // 2048            b_ih+b_hh (step 0, c=y=h=0)
#define WS_BG       231424u   // 1024*2048       packed [W_ih_y ; W_hh], pair-interleaved [(k>>1)][n][2]
#define WS_BA       2328576u  // 1024*512        packed W_attn[:, :1024], pair-interleaved [(k>>1)][m][2]
#define WS_HBUF     2852864u  // 2 * 64*512      h double buffer
#define WS_CELL     2918400u  // 64*512
#define WS_YZERO    2951168u  // 64*512          zeros (y_{-1})
// total 2983936 floats (~11.4 MiB)

// ---------- setup kernels ----------

__global__ __launch_bounds__(256) void k_zero(float* __restrict__ p, int n) {
  int i = blockIdx.x * 256 + threadIdx.x;
  if (i < n) p[i] = 0.0f;
}

// s[t*64+b] = dot(inputs[t,b,:], W_align[0,:512]) + b_align   (one wave per row)
__global__ __launch_bounds__(256) void k_align(const float* __restrict__ inputs,
                                               const float* __restrict__ W_align,
                                               const float* __restrict__ b_align,
                                               float* __restrict__ s) {
  int wid = threadIdx.x >> 5, lane = threadIdx.x & 31;
  int row = blockIdx.x * 8 + wid;  // row = t*64 + b
  const float* base = inputs + (size_t)row * 512;
  float acc = 0.0f;
  for (int i = lane; i < 512; i += 32) acc += base[i] * W_align[i];
  for (int off = 16; off; off >>= 1) acc += __shfl_xor(acc, off);
  if (lane == 0) s[row] = acc + b_align[0];
}

// softmax over T (axis 0) per batch column b; in-place on s (stride 64)
__global__ __launch_bounds__(512) void k_softmax_t(float* __restrict__ s) {
  __shared__ float red[512];
  int b = blockIdx.x, t = threadIdx.x;
  float v = s[t * 64 + b];
  red[t] = v; __syncthreads();
  for (int off = 256; off; off >>= 1) { if (t < off) red[t] = fmaxf(red[t], red[t + off]); __syncthreads(); }
  float mx = red[0]; __syncthreads();
  float e = expf(v - mx);
  red[t] = e; __syncthreads();
  for (int off = 256; off; off >>= 1) { if (t < off) red[t] += red[t + off]; __syncthreads(); }
  s[t * 64 + b] = e / red[0];
}

// ctx[b,i] = sum_t inputs[t,b,i] * wts[t,b]
__global__ __launch_bounds__(256) void k_ctx(const float* __restrict__ inputs,
                                             const float* __restrict__ wts,
                                             float* __restrict__ ctx) {
  int b = blockIdx.x;
  for (int i0 = threadIdx.x; i0 < 512; i0 += 256) {
    float acc = 0.0f;
    for (int t = 0; t < 512; ++t)
      acc += inputs[((size_t)t * 64 + b) * 512 + i0] * wts[t * 64 + b];
    ctx[b * 512 + i0] = acc;
  }
}

// Pack B operands pair-interleaved so a WMMA B-fragment is one aligned 8B load:
//   Bg flat = (k>>1)*4096 + n*2 + (k&1),  value = k<512 ? W_ih[n,512+k] : W_hh[n,k-512]
//   Ba flat = (k>>1)*1024 + m*2 + (k&1),  value = W_attn[m,k]  (k<1024)
__global__ __launch_bounds__(256) void k_pack(const float* __restrict__ W_ih,
                                              const float* __restrict__ W_hh,
                                              const float* __restrict__ W_attn,
                                              float* __restrict__ Bg,
                                              float* __restrict__ Ba) {
  int idx = blockIdx.x * 256 + threadIdx.x;
  if (idx < 1024 * 2048) {
    int khalf = idx >> 12;          // k pair index
    int rem = idx & 4095;
    int n = rem >> 1;
    int k = khalf * 2 + (rem & 1);
    Bg[idx] = (k < 512) ? W_ih[(size_t)n * 1024 + 512 + k] : W_hh[(size_t)n * 512 + (k - 512)];
  } else {
    int i2 = idx - 1024 * 2048;
    if (i2 < 1024 * 512) {
      int khalf = i2 >> 10;
      int rem = i2 & 1023;
      int m = rem >> 1;
      int k = khalf * 2 + (rem & 1);
      Ba[i2] = W_attn[(size_t)m * 1536 + k];
    }
  }
}

// gates_c[b,n] = ctx[b]·W_ih[n,:512] + b_ih[n] + b_hh[n];  attn_c[b,m] = ctx[b]·W_attn[m,1024:] + b_attn[m];
// bias0[n] = b_ih[n] + b_hh[n]
__global__ __launch_bounds__(256) void k_cpre(const float* __restrict__ ctx,
                                              const float* __restrict__ W_ih,
                                              const float* __restrict__ b_ih,
                                              const float* __restrict__ b_hh,
                                              const float* __restrict__ W_attn,
                                              const float* __restrict__ b_attn,
                                              float* __restrict__ gates_c,
                                              float* __restrict__ attn_c,
                                              float* __restrict__ bias0) {
  int idx = blockIdx.x * 256 + threadIdx.x;
  if (idx < 64 * 2048) {
    int b = idx >> 11, n = idx & 2047;
    float acc = b_ih[n] + b_hh[n];
    const float* c = ctx + b * 512;
    const float* w = W_ih + (size_t)n * 1024;
    for (int i = 0; i < 512; ++i) acc += c[i] * w[i];
    gates_c[idx] = acc;
  } else if (idx < 64 * 2048 + 64 * 512) {
    int i2 = idx - 64 * 2048;
    int b = i2 >> 9, m = i2 & 511;
    float acc = b_attn[m];
    const float* c = ctx + b * 512;
    const float* w = W_attn + (size_t)m * 1536 + 1024;
    for (int i = 0; i < 512; ++i) acc += c[i] * w[i];
    attn_c[i2] = acc;
  } else if (idx < 64 * 2048 + 64 * 512 + 2048) {
    int n = idx - (64 * 2048 + 64 * 512);
    bias0[n] = b_ih[n] + b_hh[n];
  }
}

// ---------- per-step kernels (fp32 WMMA 16x16x4) ----------
// gates = base + [y|h](64x1024) @ Bg(1024x2048), then fused LSTM pointwise.
// 32 blocks x 256 thr (8 waves); block owns gate columns j0..j0+15 in all 4 quadrants.
__global__ __launch_bounds__(256) void k_gates(const float* __restrict__ yprev,
                                               const float* __restrict__ hprev,
                                               const float* __restrict__ Bg,
                                               const float* __restrict__ base, int baseIsMat,
                                               float* __restrict__ cell,
                                               float* __restrict__ hnew) {
  __shared__ float As[64][132];     // A chunk: 64 rows x 128 K (pad->132: 16B-aligned, no bank conflicts)
  __shared__ float Gs[4][64][16];   // gate quadrants for the pointwise phase
  const int tid = threadIdx.x;
  const int lane = tid & 31, wid = tid >> 5;
  const int half = lane >> 4, l = lane & 15;
  const int q = wid >> 1;                    // gate quadrant (i,f,g,o)
  const int mt0 = (wid & 1) * 2, mt1 = mt0 + 1;
  const int j0 = blockIdx.x * 16;
  const int n = q * 512 + j0 + l;

  v8f acc0 = {}, acc1 = {};
  for (int kc = 0; kc < 1024; kc += 128) {
    __syncthreads();
    const float* src = (kc < 512) ? (yprev + kc) : (hprev + (kc - 512));
    for (int i = tid; i < 64 * 32; i += 256) {
      int row = i >> 5, c4 = (i & 31) * 4;
      *(float4*)&As[row][c4] = *(const float4*)(src + row * 512 + c4);
    }
    __syncthreads();
    for (int kk = 0; kk < 128; kk += 4) {
      int kA = kk + 2 * half;
      float2 a0 = *(const float2*)&As[mt0 * 16 + l][kA];
      float2 a1 = *(const float2*)&As[mt1 * 16 + l][kA];
      int kG = kc + kk;
      float2 bp = *(const float2*)(Bg + (size_t)(((kG >> 1) + half) * 4096 + n * 2));
      v2f bv; bv.x = bp.x; bv.y = bp.y;
      v2f av0; av0.x = a0.x; av0.y = a0.y;
      v2f av1; av1.x = a1.x; av1.y = a1.y;
      acc0 = __builtin_amdgcn_wmma_f32_16x16x4_f32(false, av0, false, bv, (short)0, acc0, false, false);
      acc1 = __builtin_amdgcn_wmma_f32_16x16x4_f32(false, av1, false, bv, (short)0, acc1, false, false);
    }
  }
  __syncthreads();
  for (int v = 0; v < 8; ++v) {
    int r0 = mt0 * 16 + v + 8 * half;
    int r1 = mt1 * 16 + v + 8 * half;
    float b0 = baseIsMat ? base[r0 * 2048 + n] : base[n];
    float b1 = baseIsMat ? base[r1 * 2048 + n] : base[n];
    Gs[q][r0][l] = acc0[v] + b0;
    Gs[q][r1][l] = acc1[v] + b1;
  }
  __syncthreads();
  for (int e = tid; e < 64 * 16; e += 256) {   // LSTM pointwise on owned 64x16 slice
    int m = e >> 4, jj = e & 15;
    float gi = Gs[0][m][jj], gf = Gs[1][m][jj], gg = Gs[2][m][jj], go = Gs[3][m][jj];
    int idx = m * 512 + j0 + jj;
    float cn = sigf(gf) * cell[idx] + sigf(gi) * tanhf(gg);
    cell[idx] = cn;
    hnew[idx] = sigf(go) * tanhf(cn);
  }
}

// z = attn_c + [y|h_new](64x1024) @ Ba(1024x512), then fused row softmax -> out[t].
// 4 blocks x 512 thr (16 waves); block owns 16 batch rows x all 512 cols.
__global__ __launch_bounds__(512) void k_attn(const float* __restrict__ yprev,
                                              const float* __restrict__ hnew,
                                              const float* __restrict__ Ba,
                                              const float* __restrict__ attn_c,
                                              float* __restrict__ out_t) {
  __shared__ float As[16][132];
  __shared__ float Zs[16][512];
  const int tid = threadIdx.x;
  const int lane = tid & 31, wid = tid >> 5;   // 16 waves, one row each for softmax
  const int half = lane >> 4, l = lane & 15;
  const int M0 = blockIdx.x * 16;
  const int n0 = wid * 32;
  const int nA = n0 + l, nB = n0 + 16 + l;

  v8f acc0 = {}, acc1 = {};
  for (int kc = 0; kc < 1024; kc += 128) {
    __syncthreads();
    const float* src = (kc < 512) ? (yprev + kc) : (hnew + (kc - 512));
    for (int i = tid; i < 16 * 32; i += 512) {
      int row = i >> 5, c4 = (i & 31) * 4;
      *(float4*)&As[row][c4] = *(const float4*)(src + (M0 + row) * 512 + c4);
    }
    __syncthreads();
    for (int kk = 0; kk < 128; kk += 4) {
      int kA = kk + 2 * half;
      float2 a = *(const float2*)&As[l][kA];
      v2f av; av.x = a.x; av.y = a.y;
      int kG = kc + kk;
      int prow = ((kG >> 1) + half) * 1024;
      float2 bp0 = *(const float2*)(Ba + (size_t)(prow + nA * 2));
      float2 bp1 = *(const float2*)(Ba + (size_t)(prow + nB * 2));
      v2f b0; b0.x = bp0.x; b0.y = bp0.y;
      v2f b1; b1.x = bp1.x; b1.y = bp1.y;
      acc0 = __builtin_amdgcn_wmma_f32_16x16x4_f32(false, av, false, b0, (short)0, acc0, false, false);
      acc1 = __builtin_amdgcn_wmma_f32_16x16x4_f32(false, av, false, b1, (short)0, acc1, false, false);
    }
  }
  __syncthreads();
  for (int v = 0; v < 8; ++v) {
    int r = v + 8 * half;
    Zs[r][nA] = acc0[v] + attn_c[(M0 + r) * 512 + nA];
    Zs[r][nB] = acc1[v] + attn_c[(M0 + r) * 512 + nB];
  }
  __syncthreads();
  {
    int r = wid;
    float mx = -3.0e38f;
    for (int c = lane; c < 512; c += 32) mx = fmaxf(mx, Zs[r][c]);
    for (int off = 16; off; off >>= 1) mx = fmaxf(mx, __shfl_xor(mx, off));
    float sum = 0.0f;
    for (int c = lane; c < 512; c += 32) { float e = expf(Zs[r][c] - mx); Zs[r][c] = e; sum += e; }
    for (int off = 16; off; off >>= 1) sum += __shfl_xor(sum, off);
    float inv = 1.0f / sum;
    for (int c = lane; c < 512; c += 32) out_t[(size_t)(M0 + r) * 512 + c] = Zs[r][c] * inv;
  }
}

extern "C" void kernel_launch(void* const* d_in, const int* in_sizes, int n_in,
                              void* d_out, int out_size, void* d_ws, size_t ws_size,
                              hipStream_t stream) {
  const float* inputs  = (const float*)d_in[0];
  const float* W_ih    = (const float*)d_in[1];
  const float* W_hh    = (const float*)d_in[2];
  const float* b_ih    = (const float*)d_in[3];
  const float* b_hh    = (const float*)d_in[4];
  const float* W_align = (const float*)d_in[5];
  const float* b_align = (const float*)d_in[6];
  const float* W_attn  = (const float*)d_in[7];
  const float* b_attn  = (const float*)d_in[8];
  float* out = (float*)d_out;
  float* ws  = (float*)d_ws;

  float* wts    = ws + WS_WTS;
  float* ctx    = ws + WS_CTX;
  float* gatesc = ws + WS_GATESC;
  float* attnc  = ws + WS_ATTNC;
  float* bias0  = ws + WS_BIAS0;
  float* Bg     = ws + WS_BG;
  float* Ba     = ws + WS_BA;
  float* Hbuf   = ws + WS_HBUF;
  float* cell   = ws + WS_CELL;
  float* yzero  = ws + WS_YZERO;

  // zero carry state every call (Hbuf both planes + cell + yzero, contiguous)
  k_zero<<<512, 256, 0, stream>>>(Hbuf, 131072);
  // attention weights & context are step-invariant (softmax shift invariance) — hoist out of loop
  k_align<<<4096, 256, 0, stream>>>(inputs, W_align, b_align, wts);
  k_softmax_t<<<64, 512, 0, stream>>>(wts);
  k_ctx<<<64, 256, 0, stream>>>(inputs, wts, ctx);
  k_pack<<<10240, 256, 0, stream>>>(W_ih, W_hh, W_attn, Bg, Ba);
  k_cpre<<<648, 256, 0, stream>>>(ctx, W_ih, b_ih, b_hh, W_attn, b_attn, gatesc, attnc, bias0);

  for (int t = 0; t < 512; ++t) {
    const float* yprev = (t == 0) ? yzero : (out + (size_t)(t - 1) * 64 * 512);
    float* hprev = Hbuf + (t & 1) * 32768;
    float* hnew  = Hbuf + ((t + 1) & 1) * 32768;
    const float* base = (t == 0) ? bias0 : gatesc;  // t=0: c=y=h=0 -> gates are just biases
    k_gates<<<32, 256, 0, stream>>>(yprev, hprev, Bg, base, (t != 0) ? 1 : 0, cell, hnew);
    k_attn<<<4, 512, 0, stream>>>(yprev, hnew, Ba, attnc, out + (size_t)t * 64 * 512);
  }
}